// TemporalPatternEncoder_81226421502238
// MI455X (gfx1250) — compile-verified
//
#include <hip/hip_runtime.h>
#include <math.h>

#define HIST_LEN 101
#define W_WIN 10
#define NUM_K 5
#define KSZ 10
#define D_DIM 8192
#define B_DIM 256
#define P_DIM 1024
#define EPSV 1e-6f

typedef float v2f __attribute__((ext_vector_type(2)));
typedef float v8f __attribute__((ext_vector_type(8)));

// ---------------------------------------------------------------------------
// Kernel 1: build `recent` (last W rows of history with row hist_idx replaced
// by input_spikes[0]) and rinv[k][d] = 1 / (sqrt(sum_w K2[k][w]*recent^2)+eps)
// ---------------------------------------------------------------------------
__global__ __launch_bounds__(256) void tpe_prep(
    const float* __restrict__ input_spikes,   // (B, D) -- row 0 used
    const float* __restrict__ spike_history,  // (HIST_LEN, D)
    const float* __restrict__ kern,           // (NUM_K, KSZ)
    const int*   __restrict__ hist_index,     // scalar
    float* __restrict__ recent,               // (W_WIN, D)
    float* __restrict__ rinv)                 // (NUM_K, D)
{
    __shared__ float K2s[NUM_K * KSZ];
    int tid = threadIdx.x;
    if (tid < NUM_K * KSZ) { float v = kern[tid]; K2s[tid] = v * v; }
    __syncthreads();

    int d = blockIdx.x * 256 + tid;           // grid sized so d < D_DIM always
    int r = hist_index[0] % HIST_LEN;

    float s[NUM_K];
#pragma unroll
    for (int k = 0; k < NUM_K; ++k) s[k] = 0.0f;

#pragma unroll
    for (int w = 0; w < W_WIN; ++w) {
        int hrow = HIST_LEN - W_WIN + w;
        float val = (hrow == r) ? input_spikes[d]
                                : spike_history[(size_t)hrow * D_DIM + d];
        recent[(size_t)w * D_DIM + d] = val;
        float v2 = val * val;
#pragma unroll
        for (int k = 0; k < NUM_K; ++k) s[k] = fmaf(K2s[k * KSZ + w], v2, s[k]);
    }
#pragma unroll
    for (int k = 0; k < NUM_K; ++k)
        rinv[(size_t)k * D_DIM + d] = 1.0f / (sqrtf(s[k]) + EPSV);
}

// ---------------------------------------------------------------------------
// Kernel 2: strength[p] = mean_{k,d} num/(rnorm*tnorm).  One block per p,
// coalesced streaming of the 10*8192 template slice (HBM-dominant phase).
// ---------------------------------------------------------------------------
__global__ __launch_bounds__(256) void tpe_strength(
    const float* __restrict__ templ,    // (P, W_WIN, D)
    const float* __restrict__ recent,   // (W_WIN, D)  (L2-resident)
    const float* __restrict__ rinv,     // (NUM_K, D)  (L2-resident)
    const float* __restrict__ kern,     // (NUM_K, KSZ)
    float* __restrict__ strength)       // (P)
{
    __shared__ float K2s[NUM_K * KSZ];
    __shared__ float red[256];
    int tid = threadIdx.x;
    int p   = blockIdx.x;
    if (tid < NUM_K * KSZ) { float v = kern[tid]; K2s[tid] = v * v; }
    __syncthreads();

    const float* tp = templ + (size_t)p * W_WIN * D_DIM;
    float acc = 0.0f;

    for (int d = tid; d < D_DIM; d += 256) {
        float t[W_WIN], rc[W_WIN];
#pragma unroll
        for (int w = 0; w < W_WIN; ++w) {
            t[w]  = tp[(size_t)w * D_DIM + d];
            rc[w] = recent[(size_t)w * D_DIM + d];
        }
#pragma unroll
        for (int k = 0; k < NUM_K; ++k) {
            float num = 0.0f, tns = 0.0f;
#pragma unroll
            for (int w = 0; w < W_WIN; ++w) {
                float kw = K2s[k * KSZ + w];
                num = fmaf(kw * rc[w], t[w], num);
                tns = fmaf(kw * t[w],  t[w], tns);
            }
            acc += num * rinv[(size_t)k * D_DIM + d] / (sqrtf(tns) + EPSV);
        }
    }

    red[tid] = acc;
    __syncthreads();
#pragma unroll
    for (int off = 128; off > 0; off >>= 1) {
        if (tid < off) red[tid] += red[tid + off];
        __syncthreads();
    }
    if (tid == 0)
        strength[p] = red[0] * (1.0f / (float)(NUM_K * D_DIM));
}

// ---------------------------------------------------------------------------
// Kernel 3: out = sigmoid(A @ Wt^T + strength)  via V_WMMA_F32_16X16X4_F32.
// One wave computes a 16(b) x 64(p) strip: 4 C accumulators share the A frag.
// A-frag layout (ISA 7.12.2, 32-bit A 16x4): lanes 0-15 -> M=lane, K={0,1};
// lanes 16-31 -> M=lane-16, K={2,3}.  -> one float2 load per matrix per step.
// ---------------------------------------------------------------------------
__global__ __launch_bounds__(256) void tpe_gemm(
    const float* __restrict__ A,         // (B_DIM, D)   input_spikes
    const float* __restrict__ Wt,        // (P_DIM, D)   pattern_weights
    const float* __restrict__ strength,  // (P_DIM)
    float* __restrict__ out)             // (B_DIM, P_DIM)
{
    int tid   = threadIdx.x;
    int lane  = tid & 31;
    int wave  = (blockIdx.x << 3) + (tid >> 5);   // 0..255
    int btile = wave >> 4;                        // 0..15
    int pgrp  = wave & 15;                        // 0..15
    int bBase = btile << 4;
    int pBase = pgrp << 6;

    int mr   = lane & 15;
    int kOff = (lane >> 4) << 1;                  // 0 (lanes<16) or 2

    const float* aP  = A  + (size_t)(bBase + mr)      * D_DIM + kOff;
    const float* b0P = Wt + (size_t)(pBase +  0 + mr) * D_DIM + kOff;
    const float* b1P = Wt + (size_t)(pBase + 16 + mr) * D_DIM + kOff;
    const float* b2P = Wt + (size_t)(pBase + 32 + mr) * D_DIM + kOff;
    const float* b3P = Wt + (size_t)(pBase + 48 + mr) * D_DIM + kOff;

    v8f c0 = {}, c1 = {}, c2 = {}, c3 = {};

    for (int d = 0; d < D_DIM; d += 4) {
        v2f a  = *(const v2f*)(aP  + d);
        v2f b0 = *(const v2f*)(b0P + d);
        v2f b1 = *(const v2f*)(b1P + d);
        v2f b2 = *(const v2f*)(b2P + d);
        v2f b3 = *(const v2f*)(b3P + d);
        c0 = __builtin_amdgcn_wmma_f32_16x16x4_f32(false, a, false, b0, (short)0, c0, false, false);
        c1 = __builtin_amdgcn_wmma_f32_16x16x4_f32(false, a, false, b1, (short)0, c1, false, false);
        c2 = __builtin_amdgcn_wmma_f32_16x16x4_f32(false, a, false, b2, (short)0, c2, false, false);
        c3 = __builtin_amdgcn_wmma_f32_16x16x4_f32(false, a, false, b3, (short)0, c3, false, false);
    }

    // C/D layout: VGPR r -> M=r (lanes 0-15) or M=r+8 (lanes 16-31), N = lane&15
    float s0 = strength[pBase +  0 + mr];
    float s1 = strength[pBase + 16 + mr];
    float s2 = strength[pBase + 32 + mr];
    float s3 = strength[pBase + 48 + mr];
    int row0 = bBase + ((lane >> 4) << 3);

#pragma unroll
    for (int rr = 0; rr < 8; ++rr) {
        size_t o = (size_t)(row0 + rr) * P_DIM + pBase + mr;
        float x;
        x = c0[rr] + s0; out[o +  0] = 1.0f / (1.0f + expf(-x));
        x = c1[rr] + s1; out[o + 16] = 1.0f / (1.0f + expf(-x));
        x = c2[rr] + s2; out[o + 32] = 1.0f / (1.0f + expf(-x));
        x = c3[rr] + s3; out[o + 48] = 1.0f / (1.0f + expf(-x));
    }
}

// ---------------------------------------------------------------------------
extern "C" void kernel_launch(void* const* d_in, const int* in_sizes, int n_in,
                              void* d_out, int out_size, void* d_ws, size_t ws_size,
                              hipStream_t stream) {
    const float* input_spikes      = (const float*)d_in[0];
    const float* spike_history     = (const float*)d_in[1];
    const float* pattern_templates = (const float*)d_in[2];
    const float* pattern_weights   = (const float*)d_in[3];
    const float* kern              = (const float*)d_in[4];
    const int*   hist_index        = (const int*)d_in[5];
    // d_in[6] (current_time) unused by the reference math.

    float* ws       = (float*)d_ws;
    float* recent   = ws;                        // 10 * 8192 floats
    float* rinv     = ws + W_WIN * D_DIM;        //  5 * 8192 floats
    float* strength = rinv + NUM_K * D_DIM;      //  1024 floats
    float* out      = (float*)d_out;

    tpe_prep<<<D_DIM / 256, 256, 0, stream>>>(input_spikes, spike_history, kern,
                                              hist_index, recent, rinv);
    tpe_strength<<<P_DIM, 256, 0, stream>>>(pattern_templates, recent, rinv,
                                            kern, strength);
    tpe_gemm<<<32, 256, 0, stream>>>(input_spikes, pattern_weights, strength, out);
}